// SGC_18433999635061
// MI455X (gfx1250) — compile-verified
//
#include <hip/hip_runtime.h>
#include <math.h>

#define D_FEAT    128
#define N_CLASSES 64

typedef float v2f __attribute__((ext_vector_type(2)));
typedef float v8f __attribute__((ext_vector_type(8)));

// ---------------------------------------------------------------------------
// 1) zero degree counters
// ---------------------------------------------------------------------------
__global__ void sgc_init_deg(unsigned int* __restrict__ deg, int n) {
    int i = blockIdx.x * blockDim.x + threadIdx.x;
    if (i < n) deg[i] = 0u;
}

// ---------------------------------------------------------------------------
// 2) deg[dst]++ over all edges
// ---------------------------------------------------------------------------
__global__ void sgc_count_deg(const int* __restrict__ dst,
                              unsigned int* __restrict__ deg, int e) {
    int i = blockIdx.x * blockDim.x + threadIdx.x;
    if (i < e) atomicAdd(&deg[dst[i]], 1u);
}

// ---------------------------------------------------------------------------
// 3) dinv = rsqrt(deg + 1)   (+1 = self loop; always > 0)
// ---------------------------------------------------------------------------
__global__ void sgc_dinv(const unsigned int* __restrict__ deg,
                         float* __restrict__ dinv, int n) {
    int i = blockIdx.x * blockDim.x + threadIdx.x;
    if (i < n) dinv[i] = rsqrtf((float)(deg[i] + 1u));
}

// ---------------------------------------------------------------------------
// 4) y = x @ W^T  via V_WMMA_F32_16X16X4_F32 (fp32 WMMA, wave32).
//    One wave per 16-node tile; 4 class tiles of 16; K loop over 128 in
//    steps of 4. Also initializes d_out = b + y * dinv^2 (self-loop + bias).
//
//    A (16x4, f32): lane l in [0,16): a.x=A[m=l][k0+0], a.y=A[m=l][k0+1]
//                   lane l+16:        a.x=A[m=l][k0+2], a.y=A[m=l][k0+3]
//    B (4x16, f32): column n = lane%16, same K split per lane half.
//    C/D (16x16, f32): acc[r] = element (m = r + 8*(lane/16), n = lane%16).
// ---------------------------------------------------------------------------
__global__ __launch_bounds__(32) void sgc_gemm_wmma(
        const float* __restrict__ x,     // [N,128]
        const float* __restrict__ W,     // [64,128] (row = class)
        const float* __restrict__ bias,  // [64]
        const float* __restrict__ dinv,  // [N]
        float* __restrict__ y,           // [N,64]
        float* __restrict__ out,         // [N,64]
        int n) {
    int node0 = blockIdx.x * 16;
    if (node0 >= n) return;

    int lane  = threadIdx.x & 31;
    int ml    = lane & 15;          // row (A) / column (B) owned by lane
    int khalf = (lane >> 4) << 1;   // 0 or 2

    const float* xrow = x + (size_t)(node0 + ml) * D_FEAT;

    for (int ct = 0; ct < 4; ++ct) {
        int c0 = ct * 16;
        const float* wrow = W + (size_t)(c0 + ml) * D_FEAT;
        v8f acc = {};
        #pragma unroll
        for (int k0 = 0; k0 < D_FEAT; k0 += 4) {
            v2f a, b;
            a.x = xrow[k0 + khalf];
            a.y = xrow[k0 + khalf + 1];
            b.x = wrow[k0 + khalf];
            b.y = wrow[k0 + khalf + 1];
            acc = __builtin_amdgcn_wmma_f32_16x16x4_f32(
                    /*neg_a=*/false, a, /*neg_b=*/false, b,
                    /*c_mod=*/(short)0, acc,
                    /*reuse_a=*/false, /*reuse_b=*/false);
        }
        int ncol  = lane & 15;
        int mbase = (lane >> 4) * 8;
        #pragma unroll
        for (int r = 0; r < 8; ++r) {
            int node = node0 + mbase + r;
            float v  = acc[r];
            float di = dinv[node];
            size_t o = (size_t)node * N_CLASSES + c0 + ncol;
            y[o]   = v;
            out[o] = bias[c0 + ncol] + v * di * di;   // self-loop + bias init
        }
    }
}

// ---------------------------------------------------------------------------
// 5) edge scatter in class space: out[dst] += y[src] * dinv[src]*dinv[dst]
//    One wave per edge; each lane handles 2 consecutive classes (float2).
// ---------------------------------------------------------------------------
__global__ void sgc_scatter(const int* __restrict__ src,
                            const int* __restrict__ dst,
                            const float* __restrict__ dinv,
                            const float* __restrict__ y,
                            float* __restrict__ out, int e) {
    int tid  = blockIdx.x * blockDim.x + threadIdx.x;
    int edge = tid >> 5;
    int lane = tid & 31;
    if (edge >= e) return;
    int s = src[edge];
    int d = dst[edge];
    float w = dinv[s] * dinv[d];
    const float2* yr = (const float2*)(y + (size_t)s * N_CLASSES);
    float2 v = yr[lane];
    float* orow = out + (size_t)d * N_CLASSES + 2 * lane;
    atomicAdd(orow,     v.x * w);
    atomicAdd(orow + 1, v.y * w);
}

// ---------------------------------------------------------------------------
// 6) in-place log_softmax over 64 classes; one wave32 per node.
// ---------------------------------------------------------------------------
__global__ void sgc_log_softmax(float* __restrict__ out, int n) {
    int wid  = threadIdx.x >> 5;
    int lane = threadIdx.x & 31;
    int node = blockIdx.x * (blockDim.x >> 5) + wid;
    if (node >= n) return;
    float* row = out + (size_t)node * N_CLASSES;
    float v0 = row[lane];
    float v1 = row[lane + 32];
    float m = fmaxf(v0, v1);
    #pragma unroll
    for (int off = 16; off > 0; off >>= 1)
        m = fmaxf(m, __shfl_xor(m, off, 32));
    float s = expf(v0 - m) + expf(v1 - m);
    #pragma unroll
    for (int off = 16; off > 0; off >>= 1)
        s += __shfl_xor(s, off, 32);
    float ls = logf(s);
    row[lane]      = v0 - m - ls;
    row[lane + 32] = v1 - m - ls;
}

// ---------------------------------------------------------------------------
extern "C" void kernel_launch(void* const* d_in, const int* in_sizes, int n_in,
                              void* d_out, int out_size, void* d_ws, size_t ws_size,
                              hipStream_t stream) {
    const float* x    = (const float*)d_in[0];
    const int*   ei   = (const int*)  d_in[1];
    const float* W    = (const float*)d_in[2];
    const float* bias = (const float*)d_in[3];
    float* out = (float*)d_out;

    int N = in_sizes[0] / D_FEAT;
    int E = in_sizes[1] / 2;
    const int* src = ei;
    const int* dst = ei + E;

    // workspace layout: [deg u32 x N][dinv f32 x N][y f32 x N*64]
    char* ws = (char*)d_ws;
    size_t algn = ((size_t)N * 4 + 255) & ~(size_t)255;
    unsigned int* deg  = (unsigned int*)ws;
    float*        dinv = (float*)(ws + algn);
    float*        y    = (float*)(ws + 2 * algn);

    sgc_init_deg   <<<(N + 255) / 256, 256, 0, stream>>>(deg, N);
    sgc_count_deg  <<<(E + 255) / 256, 256, 0, stream>>>(dst, deg, E);
    sgc_dinv       <<<(N + 255) / 256, 256, 0, stream>>>(deg, dinv, N);
    sgc_gemm_wmma  <<<(N + 15) / 16,    32, 0, stream>>>(x, W, bias, dinv, y, out, N);
    sgc_scatter    <<<((size_t)E * 32 + 255) / 256, 256, 0, stream>>>(src, dst, dinv, y, out, E);
    sgc_log_softmax<<<(N + 7) / 8,     256, 0, stream>>>(out, N);
}